// SphereConv2d_icopoint_10462540333223
// MI455X (gfx1250) — compile-verified
//
#include <hip/hip_runtime.h>
#include <math.h>

#define CC 64      // channels
#define HH 128     // input H
#define WW 256     // input W
#define HG 122880  // grid-sample rows (L*KH)
#define HD 40960   // depthwise output rows (stride 3, VALID)
#define NB 2       // batch

typedef __attribute__((ext_vector_type(2))) float v2f;
typedef __attribute__((ext_vector_type(8))) float v8f;

// ---------------------------------------------------------------------------
// Kernel 0: NCHW -> NHWC transpose of x so the 64 channel lanes of a sampling
// block read one contiguous 256B line per bilinear tap (L2-resident).
// ---------------------------------------------------------------------------
__global__ void transpose_nchw_to_nhwc(const float* __restrict__ x,
                                       float* __restrict__ xt) {
    const int xi = threadIdx.x;   // 0..255 (W)
    const int y  = blockIdx.x;    // H
    const int c  = blockIdx.y;    // C
    const int b  = blockIdx.z;    // B
    xt[((b * HH + y) * WW + xi) * CC + c] = x[((b * CC + c) * HH + y) * WW + xi];
}

// ---------------------------------------------------------------------------
// Kernel 1: fused grid_sample (bilinear, align_corners=False, zero pad)
//           + spatial attention (mean/max over C -> 3x3 conv -> sigmoid gate)
//           + depthwise 3x3 conv, stride (3,1), VALID.
// One 64-thread block per (hd, b) output row; thread = channel.
// Needs 5 sample rows (3 value rows + 1 halo above/below for the attention
// conv) x 3 columns = 15 positions, kept in LDS.
// ---------------------------------------------------------------------------
__global__ void fused_sample_attn_dw(const float* __restrict__ xt,
                                     const float* __restrict__ grid,
                                     const float* __restrict__ sa_w,
                                     const float* __restrict__ sa_b,
                                     const float* __restrict__ dw_w,
                                     const float* __restrict__ dw_b,
                                     float* __restrict__ out1) {
    const int c  = threadIdx.x;   // channel 0..63
    const int hd = blockIdx.x;    // 0..HD-1
    const int b  = blockIdx.y;    // 0..1

    __shared__ float sv[15][65];  // padded: stats loop strides rows w/o bank conflict
    __shared__ float savg[15];
    __shared__ float smax[15];
    __shared__ float satt[9];

    const float* xb = xt + (size_t)b * (HH * WW * CC);

    // ---- sample 15 positions: rows 3*hd-1 .. 3*hd+3, cols 0..2 ----
    for (int p = 0; p < 15; ++p) {
        const int ri = p / 3;          // 0..4 -> row offset -1..+3
        const int wc = p % 3;
        const int rg = 3 * hd + ri - 1;
        float val = 0.f;
        if (rg >= 0 && rg < HG) {
            const float gxn = grid[(rg * 3 + wc) * 2 + 0];
            const float gyn = grid[(rg * 3 + wc) * 2 + 1];
            const float gx = (gxn + 1.f) * (WW * 0.5f) - 0.5f;
            const float gy = (gyn + 1.f) * (HH * 0.5f) - 0.5f;
            const float fx0 = floorf(gx), fy0 = floorf(gy);
            const int ix0 = (int)fx0, iy0 = (int)fy0;
            const float wx1 = gx - fx0, wy1 = gy - fy0;
            const float wx0 = 1.f - wx1, wy0 = 1.f - wy1;
            // 4 taps with zero padding outside the image
            if (ix0 >= 0 && ix0 < WW && iy0 >= 0 && iy0 < HH)
                val += wx0 * wy0 * xb[(iy0 * WW + ix0) * CC + c];
            if (ix0 + 1 >= 0 && ix0 + 1 < WW && iy0 >= 0 && iy0 < HH)
                val += wx1 * wy0 * xb[(iy0 * WW + ix0 + 1) * CC + c];
            if (ix0 >= 0 && ix0 < WW && iy0 + 1 >= 0 && iy0 + 1 < HH)
                val += wx0 * wy1 * xb[((iy0 + 1) * WW + ix0) * CC + c];
            if (ix0 + 1 >= 0 && ix0 + 1 < WW && iy0 + 1 >= 0 && iy0 + 1 < HH)
                val += wx1 * wy1 * xb[((iy0 + 1) * WW + ix0 + 1) * CC + c];
        }
        sv[p][c] = val;
    }
    __syncthreads();

    // ---- channel mean / max per position (threads 0..14) ----
    if (c < 15) {
        float s = 0.f, m = -INFINITY;
        for (int i = 0; i < CC; ++i) {
            const float v = sv[c][i];
            s += v;
            m = fmaxf(m, v);
        }
        const int ri = c / 3;
        const int rg = 3 * hd + ri - 1;
        const bool rv = (rg >= 0 && rg < HG);   // conv zero-pads outside Hg
        savg[c] = rv ? s * (1.f / CC) : 0.f;
        smax[c] = rv ? m : 0.f;
    }
    __syncthreads();

    // ---- 3x3 attention conv + sigmoid gate (threads 0..8) ----
    if (c < 9) {
        const int r = c / 3, w = c % 3;
        float acc = sa_b[0];
        for (int kh = 0; kh < 3; ++kh) {
            const int ri = r + kh;               // stats row index (0..4)
            for (int kw = 0; kw < 3; ++kw) {
                const int ww = w + kw - 1;       // width zero-padded outside 0..2
                if (ww < 0 || ww > 2) continue;
                acc += sa_w[(0 * 3 + kh) * 3 + kw] * savg[ri * 3 + ww] +
                       sa_w[(1 * 3 + kh) * 3 + kw] * smax[ri * 3 + ww];
            }
        }
        satt[c] = 1.f + 1.f / (1.f + __expf(-acc));   // a*x + x = x*(sigmoid+1)
    }
    __syncthreads();

    // ---- depthwise 3x3, stride (3,1), VALID -> single output column ----
    float acc = dw_b[c];
    for (int r = 0; r < 3; ++r)
        for (int w = 0; w < 3; ++w)
            acc += dw_w[c * 9 + r * 3 + w] * sv[(r + 1) * 3 + w][c] * satt[r * 3 + w];

    out1[((size_t)b * HD + hd) * CC + c] = acc;
}

// ---------------------------------------------------------------------------
// Kernel 2: pointwise 64x64 conv as WMMA f32 GEMM.
//   out[b,o,hd] = sum_c pw_w[o,c] * out1[b*HD+hd, c] + pw_b[o]
// M = output channels (A = pw_w), N = 16 positions (B = out1^T), K = 64 input
// channels in 16 steps of V_WMMA_F32_16X16X4_F32 (f32 in / f32 acc — exact).
// Lanes vary along positions -> coalesced D stores along hd.
// ---------------------------------------------------------------------------
__global__ void pw_gemm_wmma(const float* __restrict__ out1,
                             const float* __restrict__ pw_w,
                             const float* __restrict__ pw_b,
                             float* __restrict__ out) {
    const int lane  = threadIdx.x & 31;
    const int wave  = threadIdx.x >> 5;
    const int ntile = blockIdx.x * 4 + wave;   // 0..5119 position tiles
    const int nl    = lane & 15;
    const int kb    = (lane >> 4) << 1;        // 0 or 2 (A/B fragment K sublane)

    const int pos = ntile * 16 + nl;           // flattened b*HD + hd
    const int b   = pos / HD;
    const int hd  = pos - b * HD;

    const float* brow = out1 + (size_t)pos * CC;   // B[k][n] = out1[pos_n][k]

    for (int ot = 0; ot < 4; ++ot) {           // 4 tiles of 16 output channels
        const int o_base = ot * 16;
        const float* arow = pw_w + (size_t)(o_base + nl) * CC; // A[m][k] = pw_w[o][k]
        v8f acc = {};
        for (int k0 = 0; k0 < CC; k0 += 4) {
            const v2f a  = *(const v2f*)(arow + k0 + kb);
            const v2f bb = *(const v2f*)(brow + k0 + kb);
            acc = __builtin_amdgcn_wmma_f32_16x16x4_f32(
                0, a, 0, bb, 0, acc, 0, 0);
        }
        // D layout: lane<16 -> M=r, N=lane; lane>=16 -> M=r+8, N=lane-16
        const int mo = (lane >> 4) << 3;
        for (int r = 0; r < 8; ++r) {
            const int o = o_base + r + mo;
            out[((size_t)b * CC + o) * HD + hd] = acc[r] + pw_b[o];
        }
    }
}

// ---------------------------------------------------------------------------
extern "C" void kernel_launch(void* const* d_in, const int* in_sizes, int n_in,
                              void* d_out, int out_size, void* d_ws, size_t ws_size,
                              hipStream_t stream) {
    const float* x    = (const float*)d_in[0];
    const float* grid = (const float*)d_in[1];
    const float* sa_w = (const float*)d_in[2];
    const float* sa_b = (const float*)d_in[3];
    const float* dw_w = (const float*)d_in[4];
    const float* dw_b = (const float*)d_in[5];
    const float* pw_w = (const float*)d_in[6];
    const float* pw_b = (const float*)d_in[7];
    float* out = (float*)d_out;

    float* xt   = (float*)d_ws;                         // 2*128*256*64 = 16 MB
    float* out1 = xt + (size_t)NB * HH * WW * CC;       // 2*40960*64   = 20 MB

    transpose_nchw_to_nhwc<<<dim3(HH, CC, NB), WW, 0, stream>>>(x, xt);
    fused_sample_attn_dw<<<dim3(HD, NB), CC, 0, stream>>>(
        xt, grid, sa_w, sa_b, dw_w, dw_b, out1);

    const int ntiles = (NB * HD) / 16;                  // 5120
    pw_gemm_wmma<<<ntiles / 4, 128, 0, stream>>>(out1, pw_w, pw_b, out);
}